// SPAModule_65360812310753
// MI455X (gfx1250) — compile-verified
//
#include <hip/hip_runtime.h>

// ---------------- problem constants ----------------
#define BB 16
#define NN 24
#define TT 256
#define CC 512
#define NBT (BB * TT)          // 4096 (b,t) rows
#define C4 (CC / 4)            // 128 float4 per row
#define NWGT 72                // 3 weights per channel n per (b,t)

typedef __attribute__((ext_vector_type(2))) float v2f;
typedef __attribute__((ext_vector_type(8))) float v8f;

struct NetParams { const float* W1; const float* b1; const float* W2; const float* b2; };

// =====================================================================
// Kernel 1: per-(b,t) softmax weights for all 8 groups -> ws,
//           plus the constant fw==1.0 bias channel (softmax over dim-1).
// Layout per (b,t) row in ws: [0..23]=S-group weight of n,
// [24..47]=T0 weight of n, [48..71]=T1/T2 weight of n.
// =====================================================================
template <int START, int LEN>
__device__ __forceinline__ void softmax_range(const float* p, float* out) {
  float m = p[START];
#pragma unroll
  for (int i = 1; i < LEN; ++i) m = fmaxf(m, p[START + i]);
  float e[LEN];
  float s = 0.f;
#pragma unroll
  for (int i = 0; i < LEN; ++i) { e[i] = __expf(p[START + i] - m); s += e[i]; }
  const float inv = 1.0f / s;
#pragma unroll
  for (int i = 0; i < LEN; ++i) out[START + i] = e[i] * inv;
}

__global__ __launch_bounds__(256) void weights_kernel(const float* __restrict__ P,
                                                      float* __restrict__ wgt,
                                                      float* __restrict__ bias_out) {
  const int bt = blockIdx.x * blockDim.x + threadIdx.x;
  if (bt >= NBT) return;
  const int b = bt >> 8;
  const int t = bt & 255;

  float p[NN];
#pragma unroll
  for (int n = 0; n < NN; ++n) p[n] = P[(b * NN + n) * TT + t];

  float w[NWGT];
  // S groups (contiguous ranges)
  softmax_range<0, 4>(p, w);
  softmax_range<4, 4>(p, w);
  softmax_range<8, 4>(p, w);
  softmax_range<12, 4>(p, w);
  softmax_range<16, 8>(p, w);
  // T groups
  softmax_range<0, 24>(p, w + 24);   // T0: all 24
  softmax_range<0, 12>(p, w + 48);   // T1: 0..11
  softmax_range<12, 12>(p, w + 48);  // T2: 12..23

  float* dst = wgt + (size_t)bt * NWGT;
#pragma unroll
  for (int i = 0; i < NWGT; ++i) dst[i] = w[i];

  // fusion softmax over a length-1 axis is exactly 1.0
  bias_out[(b * 9 + 8) * TT + t] = 1.0f;
}

// =====================================================================
// Kernel 2 (x8 instantiations): group-bias MLP via V_WMMA_F32_16X16X4_F32.
//   rows = 4096 (b,t) pairs, tile = 16 rows per wave.
//   H = Pg @ W1^T  (16 x G GEMM, K = G, done in G/4 wmma steps)
//   bias = b2 + sum_n relu(H[:,n] + b1[n]) * W2[n]
// A layout (f32 16x4): lanes 0-15 -> M, VGPR0 = K {0|2}, VGPR1 = K {1|3}
// B layout (f32 4x16): lanes&15 -> N, VGPR0 = K {0|2}, VGPR1 = K {1|3}
// D layout: VGPR r, lanes 0-15 -> M=r, lanes 16-31 -> M=r+8.
// =====================================================================
template <int G, int GSTART, int GIDX>
__global__ __launch_bounds__(32) void bias_wmma_kernel(const float* __restrict__ P,
                                                       NetParams np,
                                                       float* __restrict__ bias_out) {
  const int lane = threadIdx.x;
  const int half = lane >> 4;   // 0: K+0/K+1, 1: K+2/K+3
  const int m = lane & 15;
  const int tile = blockIdx.x;  // 0..255
  const int row = tile * 16 + m;
  const int rb = row >> 8;
  const int rt = row & 255;

  constexpr int NK = G / 4;            // G in {4,8,12,24} -> no K padding
  constexpr int NT = (G + 15) / 16;    // N tiles

  // A fragments: Pg[row, k] = P[rb, GSTART+k, rt]
  v2f afr[NK];
#pragma unroll
  for (int kk = 0; kk < NK; ++kk) {
    const int k0 = 4 * kk + 2 * half;
    afr[kk].x = P[(rb * NN + GSTART + k0) * TT + rt];
    afr[kk].y = P[(rb * NN + GSTART + k0 + 1) * TT + rt];
  }

  float acc[8];
#pragma unroll
  for (int r = 0; r < 8; ++r) acc[r] = 0.f;

#pragma unroll
  for (int nt = 0; nt < NT; ++nt) {
    const int n = nt * 16 + m;       // output neuron handled by this lane column
    const bool nv = (n < G);
    const int ni = nv ? n : 0;       // clamp address, select value (EXEC stays full)

    v2f bfr[NK];
#pragma unroll
    for (int kk = 0; kk < NK; ++kk) {
      const int k0 = 4 * kk + 2 * half;
      const float x0 = np.W1[ni * G + k0];
      const float x1 = np.W1[ni * G + k0 + 1];
      bfr[kk].x = nv ? x0 : 0.f;
      bfr[kk].y = nv ? x1 : 0.f;
    }

    v8f d = {0.f, 0.f, 0.f, 0.f, 0.f, 0.f, 0.f, 0.f};
#pragma unroll
    for (int kk = 0; kk < NK; ++kk) {
      d = __builtin_amdgcn_wmma_f32_16x16x4_f32(false, afr[kk], false, bfr[kk],
                                                (short)0, d, false, false);
    }

    const float b1v = np.b1[ni];
    const float b1n = nv ? b1v : 0.f;
    const float w2v = np.W2[ni];
    const float w2n = nv ? w2v : 0.f;
#pragma unroll
    for (int r = 0; r < 8; ++r) {
      float h = d[r] + b1n;
      h = (h > 0.f) ? h : 0.f;       // relu
      acc[r] += h * w2n;
    }
  }

  // butterfly reduce across the 16 lanes of each half (halves stay separate)
#pragma unroll
  for (int r = 0; r < 8; ++r) {
    float s = acc[r];
    s += __shfl_xor(s, 1, 32);
    s += __shfl_xor(s, 2, 32);
    s += __shfl_xor(s, 4, 32);
    s += __shfl_xor(s, 8, 32);
    acc[r] = s;
  }

  const float b2v = np.b2[0];
  if (m == 0) {
    // lane 0 holds rows M=0..7 (VGPR r -> M=r), lane 16 holds rows M=8..15
#pragma unroll
    for (int r = 0; r < 8; ++r) {
      const int rr = tile * 16 + half * 8 + r;
      bias_out[((rr >> 8) * 9 + GIDX) * TT + (rr & 255)] = acc[r] + b2v;
    }
  }
}

// =====================================================================
// Kernel 3: single streaming pass over X.
// Each block = one (b,t), each thread = one float4 chunk of C.
// 24 coalesced float4 loads -> 8 group accumulators (+fused) -> 9 stores.
// =====================================================================
__device__ __forceinline__ void fma4(float4& a, const float4 v, const float s) {
  a.x = fmaf(v.x, s, a.x);
  a.y = fmaf(v.y, s, a.y);
  a.z = fmaf(v.z, s, a.z);
  a.w = fmaf(v.w, s, a.w);
}

__global__ __launch_bounds__(C4) void stream_kernel(const float* __restrict__ X,
                                                    const float* __restrict__ wgt,
                                                    float* __restrict__ out) {
  const int bt = blockIdx.x;    // 0..4095
  const int b = bt >> 8;
  const int t = bt & 255;
  const int c4 = threadIdx.x;   // 0..127

  const float4* X4 = reinterpret_cast<const float4*>(X);
  float4* O4 = reinterpret_cast<float4*>(out);
  const float* w = wgt + (size_t)bt * NWGT;

  float4 acc[8];
#pragma unroll
  for (int j = 0; j < 8; ++j) acc[j] = make_float4(0.f, 0.f, 0.f, 0.f);

#pragma unroll
  for (int n = 0; n < NN; ++n) {
    const float4 v = X4[((size_t)(b * NN + n) * TT + t) * C4 + c4];
    const int sg = (n < 16) ? (n >> 2) : 4;   // S group of channel n
    const int tg = (n < 12) ? 6 : 7;          // T1 / T2 group of channel n
    fma4(acc[sg], v, w[n]);        // S-group weighted sum
    fma4(acc[5], v, w[24 + n]);    // T0 (all channels)
    fma4(acc[tg], v, w[48 + n]);   // T1 or T2
  }

  float4 fused = acc[0];
#pragma unroll
  for (int j = 1; j < 8; ++j) {
    fused.x += acc[j].x; fused.y += acc[j].y;
    fused.z += acc[j].z; fused.w += acc[j].w;
  }

#pragma unroll
  for (int j = 0; j < 8; ++j)
    O4[((size_t)(b * 9 + j) * TT + t) * C4 + c4] = acc[j];
  O4[((size_t)(b * 9 + 8) * TT + t) * C4 + c4] = fused;
}

// =====================================================================
// Host launch
// Input order (setup_inputs dict order, nets in insertion order W1,b1,W2,b2):
//   [0] X, [1] P,
//   [2..21]  params_S: 5 nets x (W1,b1,W2,b2)
//   [22..33] params_T: 3 nets x (W1,b1,W2,b2)
//   [34..37] fusion_params (unused: softmax over length-1 axis == 1.0)
// Output: final_features (16*9*256*512) ++ final_biases (16*9*256)
// Workspace: 4096 * 72 floats of softmax weights (~1.2 MB)
// =====================================================================
extern "C" void kernel_launch(void* const* d_in, const int* in_sizes, int n_in,
                              void* d_out, int out_size, void* d_ws, size_t ws_size,
                              hipStream_t stream) {
  (void)in_sizes; (void)n_in; (void)out_size; (void)ws_size;

  const float* X = (const float*)d_in[0];
  const float* P = (const float*)d_in[1];

  NetParams nets[8];
  int base = 2;
  for (int i = 0; i < 8; ++i) {
    nets[i].W1 = (const float*)d_in[base + 0];
    nets[i].b1 = (const float*)d_in[base + 1];
    nets[i].W2 = (const float*)d_in[base + 2];
    nets[i].b2 = (const float*)d_in[base + 3];
    base += 4;
  }

  float* out = (float*)d_out;
  float* bias_out = out + (size_t)BB * 9 * TT * CC;
  float* wgt = (float*)d_ws;

  weights_kernel<<<NBT / 256, 256, 0, stream>>>(P, wgt, bias_out);

  const int tiles = NBT / 16;  // 256
  bias_wmma_kernel<4, 0, 0><<<tiles, 32, 0, stream>>>(P, nets[0], bias_out);
  bias_wmma_kernel<4, 4, 1><<<tiles, 32, 0, stream>>>(P, nets[1], bias_out);
  bias_wmma_kernel<4, 8, 2><<<tiles, 32, 0, stream>>>(P, nets[2], bias_out);
  bias_wmma_kernel<4, 12, 3><<<tiles, 32, 0, stream>>>(P, nets[3], bias_out);
  bias_wmma_kernel<8, 16, 4><<<tiles, 32, 0, stream>>>(P, nets[4], bias_out);
  bias_wmma_kernel<24, 0, 5><<<tiles, 32, 0, stream>>>(P, nets[5], bias_out);
  bias_wmma_kernel<12, 0, 6><<<tiles, 32, 0, stream>>>(P, nets[6], bias_out);
  bias_wmma_kernel<12, 12, 7><<<tiles, 32, 0, stream>>>(P, nets[7], bias_out);

  stream_kernel<<<NBT, C4, 0, stream>>>(X, wgt, out);
}